// KerasCapsuleNetwork_71906342469709
// MI455X (gfx1250) — compile-verified
//
#include <hip/hip_runtime.h>
#include <math.h>

typedef __attribute__((ext_vector_type(2))) float v2f;
typedef __attribute__((ext_vector_type(4))) float v4f;
typedef __attribute__((ext_vector_type(8))) float v8f;

#define NB   2048
#define SS   200
#define DD   64
#define NI   4
#define SP   208          // S padded to 13 WMMA M-tiles
#define BST  68           // sBuf row stride: bank = (4*row + col) % 64 -> conflict-free frags
#define WST  72           // sW row stride: half-wave B-frag reads hit disjoint bank ranges

__global__ __launch_bounds__(256) void capsule_fused_kernel(
    const float* __restrict__ his,      // [B,S,D]
    const float* __restrict__ item_eb,  // [B,D]
    const float* __restrict__ Wb,       // [D,D]
    const float* __restrict__ Wp,       // [D,D]
    const float* __restrict__ bp,       // [D]
    const int*   __restrict__ mask,     // [B,S]
    float* __restrict__ out_ic,         // [B,NI,D]
    float* __restrict__ out_readout)    // [B,D]
{
  __shared__ float sBuf[SP * BST];      // his, overwritten in place by hat
  __shared__ float sW[DD * WST];        // W_bilinear
  __shared__ float sLogit[NI * SP];     // routing logits w
  __shared__ float sSw[NI * SP];        // masked softmax weights
  __shared__ float sCap[NI * DD];       // capsules
  __shared__ float sIc[NI * DD];        // relu projection
  __shared__ float sMaskF[SP];
  __shared__ float sN2[NI];
  __shared__ float sAtt[NI];

  const int b    = blockIdx.x;
  const int t    = threadIdx.x;
  const int lane = t & 31;
  const int wave = t >> 5;

  // ---------------- Phase 0: stage to LDS ----------------
  const float* hisB = his + (size_t)b * SS * DD;
  for (int q = t; q < SS * DD / 4; q += 256) {         // 3200 float4, coalesced
    int s = q >> 4;                                    // 16 float4 per row
    int c = (q & 15) << 2;
    *(v4f*)(&sBuf[s * BST + c]) = *(const v4f*)(hisB + q * 4);
  }
  for (int q = t; q < (SP - SS) * DD; q += 256) {      // zero pad rows 200..207
    int s = SS + q / DD;
    int c = q % DD;
    sBuf[s * BST + c] = 0.0f;
  }
  for (int q = t; q < DD * DD; q += 256) {             // W_bilinear
    sW[(q >> 6) * WST + (q & 63)] = Wb[q];
  }
  for (int q = t; q < SP; q += 256)
    sMaskF[q] = (q < SS && mask[(size_t)b * SS + q] != 0) ? 1.0f : 0.0f;
  for (int q = t; q < NI * SP; q += 256) sLogit[q] = 0.0f;
  __syncthreads();

  // -------- Phase 1: hat = his @ Wb, in place, WMMA f32 16x16x4 --------
  {
    const int row  = lane & 15;
    const int half = lane >> 4;                        // 0 | 1
    const int col  = lane & 15;
    for (int mt = wave; mt < SP / 16; mt += 8) {       // each wave owns disjoint rows
      const int m0 = mt * 16;
      v8f acc0 = {}, acc1 = {}, acc2 = {}, acc3 = {};
      for (int kk = 0; kk < DD / 4; ++kk) {
        const int kb = kk * 4 + half * 2;              // A: VGPR0=K{0|2}, VGPR1=K{1|3}
        v2f a = *(const v2f*)(&sBuf[(m0 + row) * BST + kb]);
        v2f b0, b1, b2, b3;
        b0.x = sW[kb * WST +  0 + col]; b0.y = sW[(kb + 1) * WST +  0 + col];
        b1.x = sW[kb * WST + 16 + col]; b1.y = sW[(kb + 1) * WST + 16 + col];
        b2.x = sW[kb * WST + 32 + col]; b2.y = sW[(kb + 1) * WST + 32 + col];
        b3.x = sW[kb * WST + 48 + col]; b3.y = sW[(kb + 1) * WST + 48 + col];
        acc0 = __builtin_amdgcn_wmma_f32_16x16x4_f32(false, a, false, b0, (short)0, acc0, false, false);
        acc1 = __builtin_amdgcn_wmma_f32_16x16x4_f32(false, a, false, b1, (short)0, acc1, false, false);
        acc2 = __builtin_amdgcn_wmma_f32_16x16x4_f32(false, a, false, b2, (short)0, acc2, false, false);
        acc3 = __builtin_amdgcn_wmma_f32_16x16x4_f32(false, a, false, b3, (short)0, acc3, false, false);
      }
      // write back: C/D VGPR r -> row m0 + r + 8*half, col = lane&15
      #pragma unroll
      for (int r = 0; r < 8; ++r) {
        const int rr = (m0 + r + half * 8) * BST;
        sBuf[rr +  0 + col] = acc0[r];
        sBuf[rr + 16 + col] = acc1[r];
        sBuf[rr + 32 + col] = acc2[r];
        sBuf[rr + 48 + col] = acc3[r];
      }
    }
  }
  __syncthreads();

  // ---------------- Phase 2: dynamic routing (3 iters) ----------------
  const int n_of_t = t >> 6;        // 0..3
  const int d_of_t = t & 63;        // 0..63
  for (int iter = 0; iter < 3; ++iter) {
    // (a) softmax over interests per position, then mask (reference order)
    if (t < SS) {
      float l0 = sLogit[0 * SP + t], l1 = sLogit[1 * SP + t];
      float l2 = sLogit[2 * SP + t], l3 = sLogit[3 * SP + t];
      float m  = fmaxf(fmaxf(l0, l1), fmaxf(l2, l3));
      float e0 = __expf(l0 - m), e1 = __expf(l1 - m);
      float e2 = __expf(l2 - m), e3 = __expf(l3 - m);
      float inv = 1.0f / (e0 + e1 + e2 + e3);
      float mk  = sMaskF[t];
      sSw[0 * SP + t] = e0 * inv * mk;
      sSw[1 * SP + t] = e1 * inv * mk;
      sSw[2 * SP + t] = e2 * inv * mk;
      sSw[3 * SP + t] = e3 * inv * mk;
    }
    __syncthreads();
    // (b) cap[n][d] = sum_s sw[n][s] * hat[s][d]
    {
      float acc = 0.0f;
      const float* swn = &sSw[n_of_t * SP];
      for (int s = 0; s < SS; ++s)
        acc = fmaf(swn[s], sBuf[s * BST + d_of_t], acc);
      sCap[t] = acc;
    }
    __syncthreads();
    if (t < NI) {                   // squared norms
      float n2 = 0.0f;
      for (int d = 0; d < DD; ++d) { float v = sCap[t * DD + d]; n2 = fmaf(v, v, n2); }
      sN2[t] = n2;
    }
    __syncthreads();
    {                               // squash scale
      float n2 = sN2[n_of_t];
      float scale = (n2 / (1.0f + n2)) / sqrtf(n2 + 1e-9f);
      sCap[t] *= scale;
    }
    __syncthreads();
    // (c) agreement update: w[n][s] += <hat[s], cap[n]>
    if (iter < 2) {
      for (int idx = t; idx < NI * SS; idx += 256) {
        int n = idx / SS, s = idx % SS;
        float acc = 0.0f;
        const float* capn = &sCap[n * DD];
        const float* hs   = &sBuf[s * BST];
        for (int d = 0; d < DD; ++d) acc = fmaf(hs[d], capn[d], acc);
        sLogit[n * SP + s] += acc;
      }
      __syncthreads();
    }
  }

  // ---------------- Phase 3: ic = relu(cap @ Wp + bp) ----------------
  {
    float acc = bp[d_of_t];
    const float* capn = &sCap[n_of_t * DD];
    for (int e = 0; e < DD; ++e)
      acc = fmaf(capn[e], Wp[e * DD + d_of_t], acc);
    acc = fmaxf(acc, 0.0f);
    sIc[t] = acc;
    out_ic[(size_t)b * NI * DD + t] = acc;
  }
  __syncthreads();

  // ---------------- Phase 4: attention argmax + readout ----------------
  if (t < NI) {
    float acc = 0.0f;
    const float* eb = item_eb + (size_t)b * DD;
    for (int d = 0; d < DD; ++d) acc = fmaf(sIc[t * DD + d], eb[d], acc);
    sAtt[t] = acc;                  // softmax is monotone: argmax on logits
  }
  __syncthreads();
  if (t < DD) {
    int best = 0; float bv = sAtt[0];
    #pragma unroll
    for (int n = 1; n < NI; ++n) { float v = sAtt[n]; if (v > bv) { bv = v; best = n; } }
    out_readout[(size_t)b * DD + t] = sIc[best * DD + t];
  }
}

extern "C" void kernel_launch(void* const* d_in, const int* in_sizes, int n_in,
                              void* d_out, int out_size, void* d_ws, size_t ws_size,
                              hipStream_t stream) {
  const float* his     = (const float*)d_in[0];   // item_his_emb [B,S,D]
  const float* item_eb = (const float*)d_in[1];   // [B,D]
  const float* Wb      = (const float*)d_in[2];   // [D,D]
  const float* Wp      = (const float*)d_in[3];   // [D,D]
  const float* bp      = (const float*)d_in[4];   // [D]
  const int*   mask    = (const int*)d_in[5];     // [B,S]

  float* out_ic      = (float*)d_out;                       // B*NI*D floats
  float* out_readout = (float*)d_out + (size_t)NB * NI * DD; // then B*D floats

  capsule_fused_kernel<<<NB, 256, 0, stream>>>(his, item_eb, Wb, Wp, bp, mask,
                                               out_ic, out_readout);
}